// TimeAwareMamba_11029476016768
// MI455X (gfx1250) — compile-verified
//
#include <hip/hip_runtime.h>
#include <math.h>

// Problem dims (from reference)
#define B_  16
#define L_  1024
#define IN_ 42
#define DM_ 128
#define NL_ 4
#define DI_ 256
#define NS_ 16
#define DR_ 8
#define DC_ 4
#define T_  (B_ * L_)          // 16384 tokens
#define DBLW_ (DR_ + 2 * NS_)  // 40

typedef __attribute__((ext_vector_type(2))) float v2f;
typedef __attribute__((ext_vector_type(8))) float v8f;

#if defined(__gfx1250__) && __has_builtin(__builtin_amdgcn_wmma_f32_16x16x4_f32)
#define USE_WMMA_F32 1
#else
#define USE_WMMA_F32 0
#endif

__device__ __forceinline__ float sigmoidf_(float x) { return 1.0f / (1.0f + __expf(-x)); }
__device__ __forceinline__ float siluf_(float x)    { return x * sigmoidf_(x); }
__device__ __forceinline__ float softplusf_(float x) {
    return (x > 20.0f) ? x : log1pf(__expf(x));
}

// ---------------------------------------------------------------------------
// f32 WMMA GEMM: Out[T x ldOut] (cols [0,Nact)) = Act[T x K] @ W[Nact x K]^T
//
// 128-thread block = 4 waves. Each wave owns MT 16-row tiles x NT 16-col
// tiles: MT*NT independent accumulator chains fed by MT+NT loads per k-step,
// with depth-1 software pipelining (next k-step's fragments prefetched before
// the current WMMA group) so loads overlap the matrix pipe.
//
// Out-of-range weight rows are CLAMPED, not zeroed: output column n depends
// only on B column n (held by lane n%16), and out-of-range columns are never
// stored, so garbage there is harmless. EXEC stays all-1s (required by WMMA).
//
// Fragment layout per CDNA5 ISA (wave32):
//   A 16x4 f32 : lane loads Act[row + lane%16][k0+kh .. k0+kh+1], kh = 2*(lane/16)
//   B 4x16 f32 : lane loads W  [col + lane%16][k0+kh .. k0+kh+1]
//   C/D 16x16  : VGPR r -> M = r + 8*(lane/16), N = lane%16
// K must be a multiple of 4 and >= 8.
// ---------------------------------------------------------------------------
template <int MT, int NT>
__global__ void __launch_bounds__(128)
gemm_wmma_f32(const float* __restrict__ Act, const float* __restrict__ W,
              float* __restrict__ Out, int K, int Nact, int ldOut)
{
    const int lane = threadIdx.x & 31;
    const int wv   = threadIdx.x >> 5;                 // 0..3
    const int l16  = lane & 15;
    const int kh   = (lane >> 4) << 1;                 // 0 or 2
    const int row0 = (blockIdx.x * 4 + wv) * (16 * MT);
    const int colBase = blockIdx.y * (16 * NT);

    const float* ap[MT];
    #pragma unroll
    for (int i = 0; i < MT; ++i)
        ap[i] = Act + (size_t)(row0 + i * 16 + l16) * K + kh;

    const float* wp[NT];
    #pragma unroll
    for (int j = 0; j < NT; ++j) {
        int wrow = colBase + j * 16 + l16;
        if (wrow >= Nact) wrow = Nact - 1;             // clamp (see note above)
        wp[j] = W + (size_t)wrow * K + kh;
    }

    v8f acc[MT][NT];
    #pragma unroll
    for (int i = 0; i < MT; ++i)
        #pragma unroll
        for (int j = 0; j < NT; ++j)
            acc[i][j] = (v8f){0.f,0.f,0.f,0.f,0.f,0.f,0.f,0.f};

#if USE_WMMA_F32
    // Prologue: stage k-step 0.
    v2f a0[MT], b0[NT];
    #pragma unroll
    for (int i = 0; i < MT; ++i) a0[i] = *(const v2f*)(ap[i]);
    #pragma unroll
    for (int j = 0; j < NT; ++j) b0[j] = *(const v2f*)(wp[j]);

    // Steady state: prefetch k0+4, then issue WMMA group for k0.
    #pragma unroll 2
    for (int k0 = 0; k0 < K - 4; k0 += 4) {
        v2f a1[MT], b1[NT];
        #pragma unroll
        for (int i = 0; i < MT; ++i) a1[i] = *(const v2f*)(ap[i] + k0 + 4);
        #pragma unroll
        for (int j = 0; j < NT; ++j) b1[j] = *(const v2f*)(wp[j] + k0 + 4);

        #pragma unroll
        for (int i = 0; i < MT; ++i)
            #pragma unroll
            for (int j = 0; j < NT; ++j)
                acc[i][j] = __builtin_amdgcn_wmma_f32_16x16x4_f32(
                    false, a0[i], false, b0[j], (short)0, acc[i][j], false, false);

        #pragma unroll
        for (int i = 0; i < MT; ++i) a0[i] = a1[i];
        #pragma unroll
        for (int j = 0; j < NT; ++j) b0[j] = b1[j];
    }

    // Epilogue: last k-step.
    #pragma unroll
    for (int i = 0; i < MT; ++i)
        #pragma unroll
        for (int j = 0; j < NT; ++j)
            acc[i][j] = __builtin_amdgcn_wmma_f32_16x16x4_f32(
                false, a0[i], false, b0[j], (short)0, acc[i][j], false, false);
#else
    // Scalar fallback with identical output mapping.
    const int mhi_f = (lane >> 4) * 8;
    #pragma unroll
    for (int i = 0; i < MT; ++i)
        for (int j = 0; j < NT; ++j)
            for (int r = 0; r < 8; ++r) {
                const float* ar = Act + (size_t)(row0 + i * 16 + r + mhi_f) * K;
                const float* wr = wp[j] - kh;
                float s = 0.f;
                for (int k = 0; k < K; ++k) s += ar[k] * wr[k];
                acc[i][j][r] = s;
            }
    (void)ap;
#endif

    const int mhi = (lane >> 4) * 8;
    #pragma unroll
    for (int i = 0; i < MT; ++i)
        #pragma unroll
        for (int j = 0; j < NT; ++j) {
            const int col = colBase + j * 16 + l16;
            if (col < Nact) {
                #pragma unroll
                for (int r = 0; r < 8; ++r)
                    Out[(size_t)(row0 + i * 16 + r + mhi) * ldOut + col] = acc[i][j][r];
            }
        }
}

// ---------------------------------------------------------------------------
// h[t, d] = x[t, :] @ Wi[d, :] + bi[d]    (K = 42, VALU is fine here)
// ---------------------------------------------------------------------------
__global__ void __launch_bounds__(DM_)
inproj_kernel(const float* __restrict__ x, const float* __restrict__ Wi,
              const float* __restrict__ bi, float* __restrict__ h)
{
    __shared__ float xs[IN_];
    const int t = blockIdx.x;
    const int d = threadIdx.x;             // 0..127
    if (d < IN_) xs[d] = x[(size_t)t * IN_ + d];
    __syncthreads();
    float s = bi[d];
    const float* w = Wi + (size_t)d * IN_;
    #pragma unroll
    for (int k = 0; k < IN_; ++k) s += xs[k] * w[k];
    h[(size_t)t * DM_ + d] = s;
}

// ---------------------------------------------------------------------------
// Depthwise causal conv (DC=4) over xi = xz[:, 0:DI], + bias, SiLU -> u
// ---------------------------------------------------------------------------
__global__ void __launch_bounds__(256)
conv_silu_kernel(const float* __restrict__ xz, const float* __restrict__ convW,
                 const float* __restrict__ convB, float* __restrict__ u)
{
    const int idx = blockIdx.x * blockDim.x + threadIdx.x;  // T_*DI_
    const int d  = idx & (DI_ - 1);
    const int t  = idx >> 8;
    const int b  = t >> 10;
    const int tt = t & (L_ - 1);
    float s = convB[d];
    const float* w = convW + d * DC_;
    #pragma unroll
    for (int k = 0; k < DC_; ++k) {
        const int tp = tt - (DC_ - 1) + k;
        if (tp >= 0)
            s += xz[(size_t)(b * L_ + tp) * (2 * DI_) + d] * w[k];
    }
    u[(size_t)t * DI_ + d] = siluf_(s);
}

// ---------------------------------------------------------------------------
// dt[t, d] = softplus(dbl[t, 0:DR] @ dtW[d, :] + dtB[d])
// ---------------------------------------------------------------------------
__global__ void __launch_bounds__(256)
dtproj_kernel(const float* __restrict__ dbl, const float* __restrict__ dtW,
              const float* __restrict__ dtB, float* __restrict__ dt)
{
    const int idx = blockIdx.x * blockDim.x + threadIdx.x;
    const int d = idx & (DI_ - 1);
    const int t = idx >> 8;
    float s = dtB[d];
    const float* w = dtW + d * DR_;
    const float* r = dbl + (size_t)t * DBLW_;
    #pragma unroll
    for (int k = 0; k < DR_; ++k) s += r[k] * w[k];
    dt[(size_t)t * DI_ + d] = softplusf_(s);
}

// ---------------------------------------------------------------------------
// Selective scan. One thread per (b, d) channel, NS=16 states in registers.
// dty holds dt on input and is overwritten in-place with the scan output y.
// Sequential over t (the true latency bottleneck of this model).
// ---------------------------------------------------------------------------
__global__ void __launch_bounds__(DI_)
scan_kernel(const float* __restrict__ dbl, const float* __restrict__ u,
            float* __restrict__ dty, const float* __restrict__ A_log)
{
    const int b = blockIdx.x;    // 0..15
    const int d = threadIdx.x;   // 0..255
    float Ad[NS_];
    #pragma unroll
    for (int n = 0; n < NS_; ++n) Ad[n] = -__expf(A_log[d * NS_ + n]);
    float st[NS_];
    #pragma unroll
    for (int n = 0; n < NS_; ++n) st[n] = 0.f;

    for (int t = 0; t < L_; ++t) {
        const size_t base = (size_t)(b * L_ + t);
        const float dtv = dty[base * DI_ + d];
        const float uv  = u[base * DI_ + d];
        const float dtu = dtv * uv;
        const float* bc = dbl + base * DBLW_ + DR_;   // B_t[0:16]
        const float* cc = bc + NS_;                   // C_t[0:16]
        float y = 0.f;
        #pragma unroll
        for (int n = 0; n < NS_; ++n) {
            const float e  = __expf(dtv * Ad[n]);
            const float sn = st[n] * e + dtu * bc[n];
            st[n] = sn;
            y += sn * cc[n];
        }
        dty[base * DI_ + d] = y;   // overwrite consumed dt with y
    }
}

// ---------------------------------------------------------------------------
// u[t,d] = (y[t,d] + u[t,d]*Dp[d]) * silu(z[t,d])   (in-place over u)
// ---------------------------------------------------------------------------
__global__ void __launch_bounds__(256)
gate_kernel(const float* __restrict__ dty, const float* __restrict__ xz,
            const float* __restrict__ Dp, float* __restrict__ u)
{
    const int idx = blockIdx.x * blockDim.x + threadIdx.x;  // == t*DI_ + d
    const int d = idx & (DI_ - 1);
    const int t = idx >> 8;
    const float uv = u[idx];
    const float y  = dty[idx] + uv * Dp[d];
    const float z  = xz[(size_t)t * (2 * DI_) + DI_ + d];
    u[idx] = y * siluf_(z);
}

// ---------------------------------------------------------------------------
// out[b] = h[b, L-1, :] @ Wo + bo
// ---------------------------------------------------------------------------
__global__ void final_kernel(const float* __restrict__ h, const float* __restrict__ Wo,
                             const float* __restrict__ bo, float* __restrict__ out)
{
    const int b = threadIdx.x;
    if (b >= B_) return;
    const float* hr = h + (size_t)(b * L_ + (L_ - 1)) * DM_;
    float s = bo[0];
    #pragma unroll 8
    for (int k = 0; k < DM_; ++k) s += hr[k] * Wo[k];
    out[b] = s;
}

// ---------------------------------------------------------------------------
extern "C" void kernel_launch(void* const* d_in, const int* in_sizes, int n_in,
                              void* d_out, int out_size, void* d_ws, size_t ws_size,
                              hipStream_t stream)
{
    (void)in_sizes; (void)n_in; (void)out_size; (void)ws_size;

    const float* x     = (const float*)d_in[0];
    const float* Wi    = (const float*)d_in[1];
    const float* bi    = (const float*)d_in[2];
    const float* in_W  = (const float*)d_in[3];
    const float* convW = (const float*)d_in[4];
    const float* convB = (const float*)d_in[5];
    const float* xprW  = (const float*)d_in[6];
    const float* dtW   = (const float*)d_in[7];
    const float* dtB   = (const float*)d_in[8];
    const float* Alog  = (const float*)d_in[9];
    const float* Dp    = (const float*)d_in[10];
    const float* outW  = (const float*)d_in[11];
    const float* Wo    = (const float*)d_in[12];
    const float* bo    = (const float*)d_in[13];
    float* out = (float*)d_out;

    // Workspace layout (floats): total 16384*1192 ~= 78.1 MB
    float* h   = (float*)d_ws;                    // T x 128
    float* xz  = h   + (size_t)T_ * DM_;          // T x 512
    float* u   = xz  + (size_t)T_ * 2 * DI_;      // T x 256
    float* dbl = u   + (size_t)T_ * DI_;          // T x 40
    float* dty = dbl + (size_t)T_ * DBLW_;        // T x 256 (dt, then y)

    inproj_kernel<<<T_, DM_, 0, stream>>>(x, Wi, bi, h);

    for (int i = 0; i < NL_; ++i) {
        const float* inWi  = in_W  + (size_t)i * 2 * DI_ * DM_;
        const float* cWi   = convW + (size_t)i * DI_ * DC_;
        const float* cBi   = convB + (size_t)i * DI_;
        const float* xpWi  = xprW  + (size_t)i * DBLW_ * DI_;
        const float* dtWi  = dtW   + (size_t)i * DI_ * DR_;
        const float* dtBi  = dtB   + (size_t)i * DI_;
        const float* Ali   = Alog  + (size_t)i * DI_ * NS_;
        const float* Dpi   = Dp    + (size_t)i * DI_;
        const float* oWi   = outW  + (size_t)i * DM_ * DI_;

        // xz = h @ in_W^T      [16384 x 512], K = 128; 2x4 tiles/wave
        gemm_wmma_f32<2, 4><<<dim3(T_ / 128, (2 * DI_) / 64), 128, 0, stream>>>(
            h, inWi, xz, DM_, 2 * DI_, 2 * DI_);

        // u = silu(depthwise_conv(xi) + b)
        conv_silu_kernel<<<(T_ * DI_) / 256, 256, 0, stream>>>(xz, cWi, cBi, u);

        // dbl = u @ xproj^T    [16384 x 40], K = 256; 2x3 tiles (N padded to 48)
        gemm_wmma_f32<2, 3><<<dim3(T_ / 128, 1), 128, 0, stream>>>(
            u, xpWi, dbl, DI_, DBLW_, DBLW_);

        // dt = softplus(dbl[:, :8] @ dtW^T + dtB)
        dtproj_kernel<<<(T_ * DI_) / 256, 256, 0, stream>>>(dbl, dtWi, dtBi, dty);

        // selective scan (overwrites dty with y)
        scan_kernel<<<B_, DI_, 0, stream>>>(dbl, u, dty, Ali);

        // u = (y + u*Dp) * silu(z)
        gate_kernel<<<(T_ * DI_) / 256, 256, 0, stream>>>(dty, xz, Dpi, u);

        // h = u @ out_W^T      [16384 x 128], K = 256; 2x4 tiles/wave
        gemm_wmma_f32<2, 4><<<dim3(T_ / 128, DM_ / 64), 128, 0, stream>>>(
            u, oWi, h, DI_, DM_, DM_);
    }

    final_kernel<<<1, 32, 0, stream>>>(h, Wo, bo, out);
}